// SharedCodebook_41446434407081
// MI455X (gfx1250) — compile-verified
//
#include <hip/hip_runtime.h>
#include <math.h>

typedef __attribute__((ext_vector_type(2))) float v2f;
typedef __attribute__((ext_vector_type(8))) float v8f;
typedef int b128_t __attribute__((vector_size(16)));   // matches builtin param

#define AS1 __attribute__((address_space(1)))
#define AS3 __attribute__((address_space(3)))

#define B_ROWS 32768
#define C_ROWS 8192
#define DIMS   256
#define KC     (DIMS / 4)     // 64 K-chunks of 4 for V_WMMA_F32_16X16X4_F32
#define NT     (C_ROWS / 16)  // 512 code tiles of 16
#define LDS_STRIDE 260        // 256 + 4 pad dwords -> conflict-free row access
#define BUFSZ (16 * LDS_STRIDE)

// ---------------- row-norm precompute: one wave per row --------------------
__global__ __launch_bounds__(256) void vq_norms(const float* __restrict__ feat,
                                                const float* __restrict__ cb,
                                                float* __restrict__ fn,
                                                float* __restrict__ cn) {
  int wid  = (blockIdx.x * blockDim.x + threadIdx.x) >> 5;
  int lane = threadIdx.x & 31;
  const float* src; float* dst; int row;
  if (wid < B_ROWS) { src = feat; dst = fn; row = wid; }
  else              { src = cb;   dst = cn; row = wid - B_ROWS; }
  const float4* p = (const float4*)(src + (size_t)row * DIMS) + lane * 2;
  float4 a = p[0], b = p[1];
  float s = a.x*a.x + a.y*a.y + a.z*a.z + a.w*a.w
          + b.x*b.x + b.y*b.y + b.z*b.z + b.w*b.w;
  for (int off = 16; off; off >>= 1) s += __shfl_xor(s, off, 32);
  if (lane == 0) dst[row] = s;
}

__device__ inline float st_out4(float4 q, float4 f, float4* o) {
  float dx = q.x - f.x, dy = q.y - f.y, dz = q.z - f.z, dw = q.w - f.w;
  // straight-through: f + (q - f), computed literally like the reference
  o->x = f.x + dx; o->y = f.y + dy; o->z = f.z + dz; o->w = f.w + dw;
  return dx*dx + dy*dy + dz*dz + dw*dw;
}

__device__ __forceinline__ void async_tile_copy(const float* g, float* l) {
  __builtin_amdgcn_global_load_async_to_lds_b128((AS1 b128_t*)g, (AS3 b128_t*)l, 0, 0);
  __builtin_amdgcn_global_load_async_to_lds_b128((AS1 b128_t*)g, (AS3 b128_t*)l, 16, 0);
  __builtin_amdgcn_global_load_async_to_lds_b128((AS1 b128_t*)g, (AS3 b128_t*)l, 32, 0);
  __builtin_amdgcn_global_load_async_to_lds_b128((AS1 b128_t*)g, (AS3 b128_t*)l, 48, 0);
}

// ---------------- fused GEMM + argmin + online-softmax ---------------------
__global__ __launch_bounds__(256) void vq_main(const float* __restrict__ feat,
                                               const float* __restrict__ cb,
                                               const float* __restrict__ fn,
                                               const float* __restrict__ cn,
                                               float* __restrict__ out,
                                               float* __restrict__ accum) {
  __shared__ float lds[2 * BUFSZ];
  const int tid  = threadIdx.x;
  const int wave = tid >> 5, lane = tid & 31;
  const int lo   = lane & 15, hi = lane >> 4;
  const int row0 = (blockIdx.x * 8 + wave) * 16;

  // A fragment (16 rows x 256 K), WMMA f32 layout, pre-scaled by -2 so the
  // accumulator directly produces -2*f.c ; kept resident in 128 VGPRs.
  v2f a[KC];
  {
    const float* fr = feat + (size_t)(row0 + lo) * DIMS + 2 * hi;
#pragma unroll
    for (int kc = 0; kc < KC; ++kc) {
      float2 t = *(const float2*)(fr + 4 * kc);
      a[kc].x = -2.0f * t.x; a[kc].y = -2.0f * t.y;
    }
  }
  float fnr[8];
#pragma unroll
  for (int r = 0; r < 8; ++r) fnr[r] = fn[row0 + 8 * hi + r];

  // Online-softmax state on logits l = -d.  m[r] = max logit = -min distance,
  // so the argmin shares m[] and needs no separate best-distance register.
  float m[8], s[8], u[8];
  int besti[8];
#pragma unroll
  for (int r = 0; r < 8; ++r) {
    besti[r] = 0; m[r] = -__builtin_inff(); s[r] = 0.0f; u[r] = 0.0f;
  }

  // async tile staging: thread -> (row tid>>4, 64-byte col block), 4 x b128
  const int trow = tid >> 4;
  const int tcol = (tid & 15) * 16;
  async_tile_copy(cb + (size_t)trow * DIMS + tcol,
                  &lds[trow * LDS_STRIDE + tcol]);
  asm volatile("s_wait_asynccnt 0x0" ::: "memory");
  __syncthreads();

  for (int t = 0; t < NT; ++t) {
    const int cur = (t & 1) * BUFSZ;
    if (t + 1 < NT) {  // async-copy next tile straight into the other buffer
      async_tile_copy(cb + (size_t)((t + 1) * 16 + trow) * DIMS + tcol,
                      &lds[((t + 1) & 1) * BUFSZ + trow * LDS_STRIDE + tcol]);
    }

    v8f acc = {0.f, 0.f, 0.f, 0.f, 0.f, 0.f, 0.f, 0.f};
    const float* bp = &lds[cur + lo * LDS_STRIDE + 2 * hi];
#pragma unroll
    for (int kc = 0; kc < KC; ++kc) {
      v2f b = *(const v2f*)(bp + 4 * kc);
      acc = __builtin_amdgcn_wmma_f32_16x16x4_f32(false, a[kc], false, b,
                                                  (short)0, acc, false, false);
    }

    const int j0 = t * 16;
    const float cnv = cn[j0 + lo];
#pragma unroll
    for (int r = 0; r < 8; ++r) {
      float d  = fnr[r] + cnv + acc[r];          // squared L2 distance
      float nl = -d;                             // logit
      bool p = nl > m[r];                        // strict: first index wins
      besti[r] = p ? (j0 + lo) : besti[r];
      float mn = p ? nl : m[r];
      float sc = __expf(m[r] - mn);
      float e  = __expf(nl - mn);
      s[r] = s[r] * sc + e;
      u[r] = u[r] * sc + d * e;
      m[r] = mn;
    }

    if (t + 1 < NT) asm volatile("s_wait_asynccnt 0x0" ::: "memory");
    __syncthreads();
  }

  // merge the 16 lanes that share each row (xor 1,2,4,8 stays in half-wave)
#pragma unroll
  for (int r = 0; r < 8; ++r) {
    for (int off = 1; off < 16; off <<= 1) {
      float om = __shfl_xor(m[r], off, 32);
      int   oi = __shfl_xor(besti[r], off, 32);
      float os = __shfl_xor(s[r], off, 32);
      float ou = __shfl_xor(u[r], off, 32);
      bool p = (om > m[r]) || (om == m[r] && oi < besti[r]);
      besti[r] = p ? oi : besti[r];
      float mn = fmaxf(m[r], om);
      float s0 = __expf(m[r] - mn), s1 = __expf(om - mn);
      s[r] = s[r] * s0 + os * s1;
      u[r] = u[r] * s0 + ou * s1;
      m[r] = mn;
    }
  }

  if (lo == 0) {  // lanes 0 and 16 hold rows row0+8*hi+r
    float hsum = 0.0f;
#pragma unroll
    for (int r = 0; r < 8; ++r)
      hsum += m[r] + __logf(s[r]) + u[r] / s[r];  // H = lse + E[d]
    atomicAdd(&accum[1], hsum);
  }

  // gather codebook[argmin] -> out, accumulate SSE for both losses
  float sse = 0.0f;
#pragma unroll
  for (int rr = 0; rr < 16; ++rr) {
    int idx = __shfl(besti[rr & 7], (rr >> 3) << 4, 32);
    const float* qp = cb   + (size_t)idx * DIMS + lane * 8;
    const float* fp = feat + (size_t)(row0 + rr) * DIMS + lane * 8;
    float*       op = out  + (size_t)(row0 + rr) * DIMS + lane * 8;
    float4 q0 = *(const float4*)qp, q1 = *(const float4*)(qp + 4);
    float4 f0 = *(const float4*)fp, f1 = *(const float4*)(fp + 4);
    float4 o0, o1;
    sse += st_out4(q0, f0, &o0);
    sse += st_out4(q1, f1, &o1);
    *(float4*)op = o0; *(float4*)(op + 4) = o1;
  }
  for (int off = 16; off; off >>= 1) sse += __shfl_xor(sse, off, 32);
  if (lane == 0) atomicAdd(&accum[0], sse);
}

// ---------------- scalars ---------------------------------------------------
__global__ void vq_finalize(const float* __restrict__ accum, float* __restrict__ out) {
  if (threadIdx.x == 0 && blockIdx.x == 0) {
    const size_t N = (size_t)B_ROWS * DIMS;
    float loss = (float)((double)accum[0] / (double)N);
    out[N]     = loss;                      // codebook_loss
    out[N + 1] = loss;                      // commitment_loss (same value)
    out[N + 2] = accum[1] / (float)B_ROWS;  // code_entropy
  }
}

extern "C" void kernel_launch(void* const* d_in, const int* in_sizes, int n_in,
                              void* d_out, int out_size, void* d_ws, size_t ws_size,
                              hipStream_t stream) {
  const float* feat = (const float*)d_in[0];   // (32768, 256)
  const float* cb   = (const float*)d_in[1];   // (8192, 256)
  float* out   = (float*)d_out;
  float* ws    = (float*)d_ws;
  float* accum = ws;            // [0]=SSE, [1]=entropy sum
  float* fn    = ws + 16;       // 32768 feature row norms
  float* cn    = fn + B_ROWS;   // 8192 codebook row norms

  (void)hipMemsetAsync(accum, 0, 16 * sizeof(float), stream);
  vq_norms<<<(B_ROWS + C_ROWS) / 8, 256, 0, stream>>>(feat, cb, fn, cn);
  vq_main<<<B_ROWS / 128, 256, 0, stream>>>(feat, cb, fn, cn, out, accum);
  vq_finalize<<<1, 64, 0, stream>>>(accum, out);
}